// HashGrid_39505109188916
// MI455X (gfx1250) — compile-verified
//
#include <hip/hip_runtime.h>

// Multiresolution hash-grid encoding (instant-NGP / tcnn style) for gfx1250.
// Gather-bound kernel: table (64 MB) lives in L2 (192 MB); coarse levels live
// in LDS (filled with CDNA5 async global->LDS ops); output streamed with
// non-temporal b128 stores.

typedef float v2f __attribute__((ext_vector_type(2)));
typedef float v4f __attribute__((ext_vector_type(4)));
typedef int   v2i __attribute__((ext_vector_type(2)));

// Address-space-qualified pointee types for the async-to-LDS builtin
// (clang signature: vV2i*1 V2i*3 Ii Ii  ->  (global int2*, local int2*, imm, imm))
typedef __attribute__((address_space(1))) v2i gas_v2i;
typedef __attribute__((address_space(3))) v2i las_v2i;

#define N_LEVELS      16
#define T_SIZE        (1u << 19)          // 524288 entries per level
#define TMASK         (T_SIZE - 1u)
#define PRIME_Y       2654435761u
#define BLOCK         256
#define N_LDS_LEVELS  4
#define LDS_ENTRIES   5178                // sum of per-level counts below

// scale_l = 16 * 1.5^l - 1  (exact in binary fp), res_l = ceil(scale)+1
constexpr float kScale[N_LEVELS] = {
    15.0f, 23.0f, 35.0f, 53.0f, 80.0f, 120.5f, 181.25f, 272.375f,
    409.0625f, 614.09375f, 921.640625f, 1382.9609375f, 2074.94140625f,
    3112.912109375f, 4669.8681640625f, 7005.30224609375f};
constexpr unsigned kRes[N_LEVELS] = {
    16, 24, 36, 54, 81, 122, 183, 274, 411, 616, 923, 1384, 2076, 3114, 4671, 7007};
// levels 0..9 are dense (res^2 <= T), 10..15 use the spatial hash.

// Dense corner index can reach res*(res+1) (the +0.5 voxel shift), so the LDS
// copy of level l needs res*(res+1)+1 entries.
constexpr int kLdsCnt[N_LDS_LEVELS] = {273, 601, 1333, 2971};
constexpr int kLdsOff[N_LDS_LEVELS] = {0, 273, 874, 2207};

#if __has_builtin(__builtin_amdgcn_global_load_async_to_lds_b64) && \
    __has_builtin(__builtin_amdgcn_s_wait_asynccnt)
#define USE_ASYNC_LDS 1
#else
#define USE_ASYNC_LDS 0
#endif

__global__ __launch_bounds__(BLOCK) void hashgrid_encode_kernel(
    const float* __restrict__ xin,   // [N,2]
    const v2f*   __restrict__ tab,   // [16, T] of float2
    v4f*         __restrict__ out,   // [N, 8] of float4  (= [N,32] floats)
    int n)
{
  __shared__ v2f stab[LDS_ENTRIES];  // 41424 B: levels 0..3 staged in LDS
  const int tid = threadIdx.x;

  // ---- Stage coarse dense tables into LDS ------------------------------
#pragma unroll
  for (int l = 0; l < N_LDS_LEVELS; ++l) {
    const v2f* src = tab + (size_t)l * T_SIZE;
    v2f* dst = stab + kLdsOff[l];
#if USE_ASYNC_LDS
    for (int e = tid; e < kLdsCnt[l]; e += BLOCK) {
      __builtin_amdgcn_global_load_async_to_lds_b64(
          (gas_v2i*)(unsigned long long)(src + e),
          (las_v2i*)(unsigned)(unsigned long long)(dst + e),
          /*imm offset=*/0, /*cpol=*/0);
    }
#else
    for (int e = tid; e < kLdsCnt[l]; e += BLOCK) dst[e] = src[e];
#endif
  }
#if USE_ASYNC_LDS
  __builtin_amdgcn_s_wait_asynccnt(0);
#endif
  __syncthreads();

  const int i = blockIdx.x * BLOCK + tid;
  if (i >= n) return;

  // read-once input: non-temporal
  const float px = __builtin_nontemporal_load(xin + 2 * (size_t)i);
  const float py = __builtin_nontemporal_load(xin + 2 * (size_t)i + 1);

  v4f pair;
#pragma unroll
  for (int l = 0; l < N_LEVELS; ++l) {
    const float scale = kScale[l];
    const float fx = fmaf(px, scale, 0.5f);
    const float fy = fmaf(py, scale, 0.5f);
    const float gx = floorf(fx);
    const float gy = floorf(fy);
    const float wx1 = fx - gx, wy1 = fy - gy;
    const float wx0 = 1.0f - wx1, wy0 = 1.0f - wy1;
    const unsigned cx = (unsigned)gx;
    const unsigned cy = (unsigned)gy;

    unsigned i00, i01, i10, i11;
    if (l < 10) {
      // dense grid: idx = cx + cy*res ; provably < T for all dense levels
      const unsigned res = kRes[l];
      const unsigned r0 = cx + cy * res;
      i00 = r0;            i10 = r0 + 1u;
      i01 = r0 + res;      i11 = r0 + res + 1u;
    } else {
      // tcnn 2D spatial hash: (cx*1) ^ (cy*PRIME), masked to T
      const unsigned h0 = cy * PRIME_Y;
      const unsigned h1 = h0 + PRIME_Y;   // (cy+1)*PRIME (mod 2^32)
      i00 = (cx        ^ h0) & TMASK;
      i10 = ((cx + 1u) ^ h0) & TMASK;
      i01 = (cx        ^ h1) & TMASK;
      i11 = ((cx + 1u) ^ h1) & TMASK;
    }

    v2f t00, t01, t10, t11;
    if (l < N_LDS_LEVELS) {
      const v2f* tl = stab + kLdsOff[l];        // ds_load_b64 gathers
      t00 = tl[i00]; t10 = tl[i10]; t01 = tl[i01]; t11 = tl[i11];
    } else {
      const v2f* tl = tab + (size_t)l * T_SIZE; // global_load_b64, L2-resident
      t00 = tl[i00]; t10 = tl[i10]; t01 = tl[i01]; t11 = tl[i11];
    }

    v2f f = t00 * (wx0 * wy0);
    f += t01 * (wx0 * wy1);
    f += t10 * (wx1 * wy0);
    f += t11 * (wx1 * wy1);

    if ((l & 1) == 0) {
      pair.x = f.x; pair.y = f.y;
    } else {
      pair.z = f.x; pair.w = f.y;
      // streamed write-once output: non-temporal b128 store
      __builtin_nontemporal_store(pair, out + (size_t)i * 8 + (l >> 1));
    }
  }
}

extern "C" void kernel_launch(void* const* d_in, const int* in_sizes, int n_in,
                              void* d_out, int out_size, void* d_ws, size_t ws_size,
                              hipStream_t stream) {
  const float* x   = (const float*)d_in[0];   // [N,2] f32
  const v2f*   tab = (const v2f*)d_in[1];     // [16, 524288, 2] f32
  v4f*         out = (v4f*)d_out;             // [N, 32] f32
  const int n = in_sizes[0] / 2;
  const int grid = (n + BLOCK - 1) / BLOCK;
  hashgrid_encode_kernel<<<grid, BLOCK, 0, stream>>>(x, tab, out, n);
}